// ConstructAdjMatrix_4002909520738
// MI455X (gfx1250) — compile-verified
//
#include <hip/hip_runtime.h>
#include <hip/hip_bf16.h>

#define NC 6144   // N_CELL
#define ND 2048   // N_DRUG
#define NT 8192   // NC + ND

typedef __attribute__((ext_vector_type(2))) float v2f;
typedef __attribute__((ext_vector_type(4))) float v4f;
typedef __attribute__((ext_vector_type(8))) float v8f;

// ---------------------------------------------------------------------------
// Row sums of M via V_WMMA_F32_16X16X4_F32:  D = A(16x4) * ones(4x16) + C.
// Every column of D equals the 4-wide partial row sum, accumulated over K.
// One wave (32 threads) handles 16 rows. A-frag layout (ISA 7.12.2):
//   lane L<16 : row L,    K = {k, k+1}   (VGPR0, VGPR1)
//   lane L>=16: row L-16, K = {k+2, k+3}
// ---------------------------------------------------------------------------
__global__ __launch_bounds__(32) void rowsum_wmma_kernel(const float* __restrict__ M,
                                                         float* __restrict__ dvec) {
  const int lane    = threadIdx.x;
  const int rowbase = blockIdx.x * 16;
  const int row     = rowbase + (lane & 15);
  const int ksub    = (lane >> 4) << 1;
  const float* p = M + (size_t)row * ND + ksub;

  v2f b = {1.0f, 1.0f};                       // ones matrix (layout-invariant)
  v8f c0 = {}; v8f c1 = {}; v8f c2 = {}; v8f c3 = {};
  for (int k = 0; k < ND; k += 16) {
    v2f a0 = *(const v2f*)(p + k);
    v2f a1 = *(const v2f*)(p + k + 4);
    v2f a2 = *(const v2f*)(p + k + 8);
    v2f a3 = *(const v2f*)(p + k + 12);
    c0 = __builtin_amdgcn_wmma_f32_16x16x4_f32(false, a0, false, b, (short)0, c0, false, false);
    c1 = __builtin_amdgcn_wmma_f32_16x16x4_f32(false, a1, false, b, (short)0, c1, false, false);
    c2 = __builtin_amdgcn_wmma_f32_16x16x4_f32(false, a2, false, b, (short)0, c2, false, false);
    c3 = __builtin_amdgcn_wmma_f32_16x16x4_f32(false, a3, false, b, (short)0, c3, false, false);
  }
  v8f c = (c0 + c1) + (c2 + c3);
  // C/D layout: VGPR r, lanes 0-15 -> row rowbase+r ; lanes 16-31 -> rowbase+8+r
  if (lane == 0) {
#pragma unroll
    for (int r = 0; r < 8; ++r) dvec[rowbase + r] = rsqrtf(1.0f + c[r]);
  } else if (lane == 16) {
#pragma unroll
    for (int r = 0; r < 8; ++r) dvec[rowbase + 8 + r] = rsqrtf(1.0f + c[r]);
  }
}

// ---------------------------------------------------------------------------
// Column sums of M (deterministic LDS reduction; lanes vary along columns so
// every global load is a contiguous 128B per wave-row). 64 blocks x 256 thr.
// ---------------------------------------------------------------------------
__global__ __launch_bounds__(256) void colsum_kernel(const float* __restrict__ M,
                                                     float* __restrict__ dvec) {
  __shared__ float red[8][32];
  const int tx  = threadIdx.x & 31;
  const int ty  = threadIdx.x >> 5;
  const int col = blockIdx.x * 32 + tx;
  float s = 0.0f;
  for (int r = ty; r < NC; r += 8)
    s += M[(size_t)r * ND + col];
  red[ty][tx] = s;
  __syncthreads();
  if (ty == 0) {
    float t = 0.0f;
#pragma unroll
    for (int k = 0; k < 8; ++k) t += red[k][tx];
    dvec[NC + col] = rsqrtf(1.0f + t);
  }
}

// ---------------------------------------------------------------------------
// Zero-fill the two diagonal blocks with the 1 + d^2 diagonal folded into the
// same float4 non-temporal store (output is write-once: keep it out of L2).
// Covers NC*NC/4 + ND*ND/4 = 10,485,760 float4 stores.
// ---------------------------------------------------------------------------
__global__ __launch_bounds__(256) void fill_diag_blocks_kernel(const float* __restrict__ dvec,
                                                               float* __restrict__ out) {
  const unsigned T1 = (unsigned)NC * NC / 4u;   // top-left float4 count
  unsigned idx = blockIdx.x * 256u + threadIdx.x;
  int r, c, base;
  if (idx < T1) {                    // top-left block, global coords == block coords
    r = (int)(idx / (NC / 4u));
    c = (int)(idx - (unsigned)r * (NC / 4u)) * 4;
    base = 0;
  } else {                           // bottom-right block, offset by NC
    idx -= T1;
    r = (int)(idx / (ND / 4u));
    c = (int)(idx - (unsigned)r * (ND / 4u)) * 4;
    base = NC;
  }
  v4f v = {0.0f, 0.0f, 0.0f, 0.0f};
  const unsigned dl = (unsigned)(r - c);
  if (dl < 4u) {                     // diagonal element lands in this float4
    const float dr = dvec[base + r];
    v[dl] = 1.0f + dr * dr;
  }
  float* dst = out + (size_t)(base + r) * NT + (base + c);
  __builtin_nontemporal_store(v, (v4f*)dst);
}

// ---------------------------------------------------------------------------
// Top-right block:  out[r, NC+j] = d[r] * M[r,j] * d[NC+j].  Fully coalesced
// float4 load of M (L2-resident) + float4 NT store.  6144*512 threads.
// ---------------------------------------------------------------------------
__global__ __launch_bounds__(256) void fill_topright_kernel(const float* __restrict__ M,
                                                            const float* __restrict__ dvec,
                                                            float* __restrict__ out) {
  const unsigned idx = blockIdx.x * 256u + threadIdx.x;
  const int r  = (int)(idx >> 9);           // / (ND/4)
  const int j  = (int)(idx & 511u) * 4;
  v4f m  = *(const v4f*)(M + (size_t)r * ND + j);
  v4f dd = *(const v4f*)(dvec + NC + j);
  const float dr = dvec[r];
  v4f v = dr * m * dd;
  float* dst = out + (size_t)r * NT + NC + j;
  __builtin_nontemporal_store(v, (v4f*)dst);
}

// ---------------------------------------------------------------------------
// Bottom-left block (transpose of top-right) via 32x33 LDS tile so both the
// M read and the output write stay coalesced. grid (NC/32, ND/32).
//   out[NC+j, c] = d[NC+j] * M[c,j] * d[c]
// ---------------------------------------------------------------------------
__global__ __launch_bounds__(256) void fill_botleft_kernel(const float* __restrict__ M,
                                                           const float* __restrict__ dvec,
                                                           float* __restrict__ out) {
  __shared__ float tile[32][33];
  const int tx = threadIdx.x & 31;
  const int ty = threadIdx.x >> 5;          // 0..7
  const int c0 = blockIdx.x * 32;           // along NC
  const int j0 = blockIdx.y * 32;           // along ND
#pragma unroll
  for (int yy = 0; yy < 32; yy += 8)
    tile[ty + yy][tx] = M[(size_t)(c0 + ty + yy) * ND + j0 + tx];
  __syncthreads();
#pragma unroll
  for (int yy = 0; yy < 32; yy += 8) {
    const int j = j0 + ty + yy;
    const int c = c0 + tx;
    const float v = dvec[NC + j] * tile[tx][ty + yy] * dvec[c];
    __builtin_nontemporal_store(v, out + (size_t)(NC + j) * NT + c);
  }
}

extern "C" void kernel_launch(void* const* d_in, const int* in_sizes, int n_in,
                              void* d_out, int out_size, void* d_ws, size_t ws_size,
                              hipStream_t stream) {
  const float* M = (const float*)d_in[0];   // [NC, ND] f32
  float* out  = (float*)d_out;              // [NT, NT] f32
  float* dvec = (float*)d_ws;               // [NT] f32 scratch: D^{-1/2}

  // Degree vector (rows via WMMA, cols via LDS reduction)
  rowsum_wmma_kernel<<<NC / 16, 32, 0, stream>>>(M, dvec);
  colsum_kernel<<<ND / 32, 256, 0, stream>>>(M, dvec);

  // Output blocks (all non-temporal, write-once)
  fill_diag_blocks_kernel<<<(NC * (NC / 4) + ND * (ND / 4)) / 256, 256, 0, stream>>>(dvec, out);
  fill_topright_kernel<<<NC * (ND / 4) / 256, 256, 0, stream>>>(M, dvec, out);
  fill_botleft_kernel<<<dim3(NC / 32, ND / 32), 256, 0, stream>>>(M, dvec, out);
}